// AutoLevelNonDifferentiable_52785148068105
// MI455X (gfx1250) — compile-verified
//
#include <hip/hip_runtime.h>

#define BATCH    16
#define HW       1048576           // 1024*1024 pixels per channel plane
#define NBINS    4096
#define HBPB     64                // histogram blocks per batch
#define ABPB     96                // apply blocks per batch
#define BLKPT_Q  0.01f
#define WHTPT_Q  0.99f
#define MAX_MULT 1.5f

typedef float f4  __attribute__((ext_vector_type(4)));
typedef float v2f __attribute__((ext_vector_type(2)));
typedef float v8f __attribute__((ext_vector_type(8)));

#if __has_builtin(__builtin_amdgcn_wmma_f32_16x16x4_f32)
#define USE_WMMA_F32 1
#else
#define USE_WMMA_F32 0
#endif

// ---- CDNA5 async global->LDS copy (inline asm; portable across toolchains) ----
// vdst VGPR = LDS byte offset (low 32 bits of generic shared pointer),
// vaddr VGPR = 32-bit byte offset, saddr = 64-bit base (GVS mode).
__device__ __forceinline__ void async_cp_b128(unsigned long long gbase,
                                              unsigned gByteOff,
                                              unsigned ldsByteOff) {
  asm volatile("global_load_async_to_lds_b128 %0, %1, %2 offset:0"
               :: "v"(ldsByteOff), "v"(gByteOff), "s"(gbase)
               : "memory");
}
__device__ __forceinline__ void wait_async_le(int n) {
  if (n == 0) asm volatile("s_wait_asynccnt 0x0" ::: "memory");
  else        asm volatile("s_wait_asynccnt 0x1" ::: "memory");
}

// ---------------- Pass 1: luminance histogram (WMMA f32 16x16x4) ----------------
__global__ __launch_bounds__(256) void hist_kernel(const float* __restrict__ img,
                                                   const float* __restrict__ mat,
                                                   unsigned* __restrict__ hist) {
  __shared__ unsigned lhist[NBINS];
  const int tid = threadIdx.x;
  for (int i = tid; i < NBINS; i += 256) lhist[i] = 0u;
  __syncthreads();

  const int b = blockIdx.x / HBPB;
  const int s = blockIdx.x % HBPB;
  const int pixPerBlock = HW / HBPB;          // 16384
  const int wave = tid >> 5;
  const int lane = tid & 31;
  const bool lo  = lane < 16;

  const float w0 = mat[0], w1 = mat[1], w2 = mat[2];
  const float* rp = img + (size_t)b * 3 * HW;
  const float* gp = rp + HW;
  const float* bp = rp + 2 * HW;

#if USE_WMMA_F32
  // A (16x4 f32): VGPR0 = K0 (lanes 0-15) / K2 (lanes 16-31); VGPR1 = K1 / K3.
  v2f a; a.x = lo ? w0 : w2; a.y = lo ? w1 : 0.0f;
#endif

  const int base0 = s * pixPerBlock;
  for (int it = 0; it < pixPerBlock; it += 8 * 64) {   // 512 px per block-iter
    const int p = base0 + it + wave * 64 + ((lane & 15) << 2);
    f4 r4 = *(const f4*)(rp + p);
    f4 g4 = *(const f4*)(gp + p);
    f4 b4 = *(const f4*)(bp + p);
#pragma unroll
    for (int j = 0; j < 4; ++j) {
      float y;
#if USE_WMMA_F32
      // B (4x16 f32): VGPR0 = row K0 (lanes 0-15) / K2 (lanes 16-31); VGPR1 = K1 / K3.
      v2f bv; bv.x = lo ? r4[j] : b4[j]; bv.y = lo ? g4[j] : 0.0f;
      v8f c = {0.f, 0.f, 0.f, 0.f, 0.f, 0.f, 0.f, 0.f};
      v8f d = __builtin_amdgcn_wmma_f32_16x16x4_f32(
          false, a, false, bv, (short)0, c, false, false);
      y = d[0];   // D VGPR0: Y of pixel (p+j), duplicated in both lane halves
#else
      y = fmaf(w0, r4[j], fmaf(w1, g4[j], w2 * b4[j]));
#endif
      int bin = (int)(y * (float)NBINS);
      bin = bin < 0 ? 0 : (bin > NBINS - 1 ? NBINS - 1 : bin);
      atomicAdd(&lhist[bin], 1u);
    }
  }
  __syncthreads();
  unsigned* gh = hist + (size_t)b * NBINS;
  for (int i = tid; i < NBINS; i += 256) {
    unsigned v = lhist[i];
    if (v) atomicAdd(&gh[i], v);
  }
}

// ---------------- Pass 2: CDF walk -> (blkpt, mult) per batch ----------------
__global__ void scan_kernel(const unsigned* __restrict__ hist,
                            float* __restrict__ params) {
  const int b = threadIdx.x;
  if (b >= BATCH) return;
  const unsigned* h = hist + (size_t)b * NBINS;
  float total = 0.f;
  for (int i = 0; i < NBINS; ++i) total += (float)h[i];
  const float scale = total / (float)HW;  // 2.0 on both paths (each pixel binned twice)
  const float tB = BLKPT_Q * (float)(HW - 1) * scale;
  const float tW = WHTPT_Q * (float)(HW - 1) * scale;
  float blk = 0.f, wht = 1.f, cum = 0.f;
  bool gotB = false, gotW = false;
  for (int i = 0; i < NBINS && !(gotB && gotW); ++i) {
    const float c = (float)h[i];
    const float nxt = cum + c;
    if (!gotB && nxt > tB) { blk = ((float)i + (c > 0.f ? (tB - cum) / c : 0.f)) * (1.0f / NBINS); gotB = true; }
    if (!gotW && nxt > tW) { wht = ((float)i + (c > 0.f ? (tW - cum) / c : 0.f)) * (1.0f / NBINS); gotW = true; }
    cum = nxt;
  }
  const float mult = fminf(1.0f / (wht - blk), MAX_MULT);
  params[2 * b]     = blk;
  params[2 * b + 1] = mult;
}

// ------- Pass 3: apply, async global->LDS double-buffered, NT stores -------
__global__ __launch_bounds__(256) void apply_kernel(const float* __restrict__ img,
                                                    const float* __restrict__ params,
                                                    float* __restrict__ out) {
  __shared__ f4 tile[2][256];
  const int b = blockIdx.x / ABPB;
  const int s = blockIdx.x % ABPB;
  const float blk  = params[2 * b];
  const float mult = params[2 * b + 1];

  const size_t batchOff = (size_t)b * 3 * HW;
  const unsigned long long src = (unsigned long long)(img + batchOff);
  f4* dst = (f4*)(out + batchOff);

  const int nv4      = 3 * HW / 4;        // 786432 f4 per batch
  const int perBlock = nv4 / ABPB;        // 8192
  const int iters    = perBlock / 256;    // 32
  const int tid  = threadIdx.x;
  const int base = s * perBlock + tid;

  const unsigned ldsOff0 = (unsigned)(unsigned long long)(&tile[0][tid]);
  const unsigned ldsOff1 = (unsigned)(unsigned long long)(&tile[1][tid]);

  // prologue: DMA iteration 0 into buffer 0
  async_cp_b128(src, (unsigned)(base * 16u), ldsOff0);
  for (int it = 0; it < iters; ++it) {
    const int buf = it & 1;
    if (it + 1 < iters) {
      async_cp_b128(src, (unsigned)((base + (it + 1) * 256) * 16u),
                    (it + 1) & 1 ? ldsOff1 : ldsOff0);
      wait_async_le(1);   // in-order: load `it` is complete
    } else {
      wait_async_le(0);
    }
    f4 v = tile[buf][tid];  // each lane reads only its own DMA'd slot
    f4 o;
#pragma unroll
    for (int j = 0; j < 4; ++j)
      o[j] = fminf(fmaxf((v[j] - blk) * mult, 0.0f), 1.0f);
    __builtin_nontemporal_store(o, &dst[base + it * 256]); // keep input L2-resident
  }
}

extern "C" void kernel_launch(void* const* d_in, const int* in_sizes, int n_in,
                              void* d_out, int out_size, void* d_ws, size_t ws_size,
                              hipStream_t stream) {
  const float* img = (const float*)d_in[0];
  const float* mat = (const float*)d_in[1];   // 3x3 row-major; row 0 = Y weights
  float* out = (float*)d_out;

  unsigned* hist = (unsigned*)d_ws;
  float* params  = (float*)((char*)d_ws + (size_t)BATCH * NBINS * sizeof(unsigned));

  hipMemsetAsync(hist, 0, (size_t)BATCH * NBINS * sizeof(unsigned), stream);
  hist_kernel<<<BATCH * HBPB, 256, 0, stream>>>(img, mat, hist);
  scan_kernel<<<1, 32, 0, stream>>>(hist, params);
  apply_kernel<<<BATCH * ABPB, 256, 0, stream>>>(img, params, out);
}